// FeatureDecorrelation_58402965291150
// MI455X (gfx1250) — compile-verified
//
#include <hip/hip_runtime.h>
#include <math.h>

typedef float v2f __attribute__((ext_vector_type(2)));
typedef float v8f __attribute__((ext_vector_type(8)));

#define D 128
#define LDS_STRIDE 132          // pad 128 -> 132 dwords: rows r and r+2 hit different banks
#define CHUNK 128               // rows of F staged in LDS per iteration (128*132*4 = 66 KB)
#define CHUNKS_PER_BLOCK 4
#define ROWS_PER_BLOCK (CHUNK * CHUNKS_PER_BLOCK)

// ---------------------------------------------------------------------------
// Kernel 1: symmetric partial Gram, upper-triangle tile pairs only.
// 8 waves; wave w computes tiles (w, (w+d) mod 8), d = 0..4, via
// v_wmma_f32_16x16x4_f32; d=4 is duplicated by waves w and w+4, so only
// waves 0-3 write it.  36 unique tile-pairs instead of 64 full tiles.
// ---------------------------------------------------------------------------
__launch_bounds__(256)
__global__ void FeatureDecorrelation_gram_kernel(const float* __restrict__ F,
                                                 float* __restrict__ G) {
    __shared__ __align__(16) float lds[CHUNK * LDS_STRIDE];

    const int tid  = threadIdx.x;
    const int lane = tid & 31;
    const int col  = lane & 15;                       // M/N position in 16-wide tile
    const int hi   = lane >> 4;                       // 0 -> K=0,1 ; 1 -> K=2,3
    const int w    = __builtin_amdgcn_readfirstlane(tid >> 5);   // uniform wave id
    const int j0   = w * 16;                          // A-row tile base (scalar)

    v8f acc[5];
    #pragma unroll
    for (int d = 0; d < 5; ++d) { v8f z = {}; acc[d] = z; }

    const long rowBase = (long)blockIdx.x * ROWS_PER_BLOCK;

    for (int c = 0; c < CHUNKS_PER_BLOCK; ++c) {
        // --- cooperative load: CHUNK*128 contiguous floats -> padded LDS ---
        const float* srcf = F + (rowBase + (long)c * CHUNK) * D;
        const float4* src = (const float4*)srcf;
        #pragma unroll
        for (int i = 0; i < (CHUNK * (D / 4)) / 256; ++i) {
            int idx = tid + i * 256;        // float4 index
            int row = idx >> 5;             // 32 float4 per row
            int c4  = idx & 31;
            float4 v = src[idx];
            *(float4*)&lds[row * LDS_STRIDE + c4 * 4] = v;
        }
        __syncthreads();

        // prefetch next chunk (one 256B line per thread covers the 64 KB chunk)
        if (c + 1 < CHUNKS_PER_BLOCK)
            __builtin_prefetch(srcf + (size_t)CHUNK * D + (size_t)tid * 64, 0, 0);

        // --- K loop: 4 rows of the chunk per WMMA ---
        #pragma unroll 2
        for (int t = 0; t < CHUNK / 4; ++t) {
            const int r0 = (t * 4 + hi * 2) * LDS_STRIDE;
            const int r1 = r0 + LDS_STRIDE;

            v2f a;                               // A: rows j0..j0+15 of G
            a.x = lds[r0 + j0 + col];
            a.y = lds[r1 + j0 + col];

            // d = 0: B fragment is identical to A (same columns, same layout)
            acc[0] = __builtin_amdgcn_wmma_f32_16x16x4_f32(
                false, a, false, a, (short)0, acc[0], false, false);

            #pragma unroll
            for (int d = 1; d < 5; ++d) {
                const int lbase = ((w + d) & 7) * 16;   // scalar
                v2f b;
                b.x = lds[r0 + lbase + col];
                b.y = lds[r1 + lbase + col];
                acc[d] = __builtin_amdgcn_wmma_f32_16x16x4_f32(
                    false, a, false, b, (short)0, acc[d], false, false);
            }
        }
        __syncthreads();
    }

    // --- merge block partials into canonical (upper-tile) positions ---
    // C layout: VGPR r, lanes 0-15 -> M=r, lanes 16-31 -> M=r+8.
    #pragma unroll
    for (int d = 0; d < 5; ++d) {
        if (d == 4 && w >= 4) continue;      // duplicated pair: only waves 0-3 write
        const int lt = (w + d) & 7;          // scalar l-tile
        const int l0 = lt * 16;
        if (lt >= w) {                       // already upper: direct write
            #pragma unroll
            for (int r = 0; r < 8; ++r) {
                int row = j0 + r + hi * 8;
                unsafeAtomicAdd(&G[row * D + l0 + col], acc[d][r]);
            }
        } else {                             // wrapped: write transposed position
            #pragma unroll
            for (int r = 0; r < 8; ++r) {
                int row = j0 + r + hi * 8;
                unsafeAtomicAdd(&G[(l0 + col) * D + row], acc[d][r]);
            }
        }
    }
}

// ---------------------------------------------------------------------------
// Kernel 2: out = sum_l ( logsumexp_j(G[j,l]/TAU2) - G[l,l]/TAU2 )
// G holds only canonical tiles (tile(j) <= tile(l)); read symmetric.
// ---------------------------------------------------------------------------
__global__ void FeatureDecorrelation_lse_kernel(const float* __restrict__ G,
                                                float* __restrict__ out) {
    __shared__ float red[D];
    const int l  = threadIdx.x;
    const int tl = l >> 4;

    float m = -INFINITY;
    for (int j = 0; j < D; ++j) {
        float g = ((j >> 4) <= tl) ? G[j * D + l] : G[l * D + j];
        m = fmaxf(m, g * 0.5f);                 // 1/TAU2 = 0.5
    }
    float s = 0.0f;
    for (int j = 0; j < D; ++j) {
        float g = ((j >> 4) <= tl) ? G[j * D + l] : G[l * D + j];
        s += expf(g * 0.5f - m);
    }
    red[l] = m + logf(s) - G[l * D + l] * 0.5f;
    __syncthreads();

    for (int off = D / 2; off > 0; off >>= 1) {
        if (l < off) red[l] += red[l + off];
        __syncthreads();
    }
    if (l == 0) out[0] = red[0];
}

extern "C" void kernel_launch(void* const* d_in, const int* in_sizes, int n_in,
                              void* d_out, int out_size, void* d_ws, size_t ws_size,
                              hipStream_t stream) {
    const float* F = (const float*)d_in[0];
    float* gram    = (float*)d_ws;              // 128*128 fp32 = 64 KB accumulator
    float* out     = (float*)d_out;
    const int N    = in_sizes[0] / D;           // 262144
    const int nblk = N / ROWS_PER_BLOCK;        // 512

    hipMemsetAsync(d_ws, 0, D * D * sizeof(float), stream);
    FeatureDecorrelation_gram_kernel<<<nblk, 256, 0, stream>>>(F, gram);
    FeatureDecorrelation_lse_kernel<<<1, D, 0, stream>>>(gram, out);
}